// GCNLinkPredictor_68779606278430
// MI455X (gfx1250) — compile-verified
//
#include <hip/hip_runtime.h>
#include <hip/hip_bf16.h>

typedef __attribute__((ext_vector_type(2))) float v2f;
typedef __attribute__((ext_vector_type(8))) float v8f;

#define D_IN   256
#define D_HID  64
#define D_EMB  32

// ---------------------------------------------------------------------------
// Zero-fill, float4-vectorized (accumulators re-zeroed every launch so the
// graph replay is deterministic). n4 = number of float4 elements.
// ---------------------------------------------------------------------------
__global__ __launch_bounds__(256) void fill_zero4(float4* __restrict__ p, long long n4) {
    long long i = (long long)blockIdx.x * blockDim.x + threadIdx.x;
    if (i < n4) p[i] = make_float4(0.f, 0.f, 0.f, 0.f);
}

// ---------------------------------------------------------------------------
// GEMM1: h[N,64] = x[N,256] @ W1[256,64]  (fp32 WMMA 16x16x4)
// block = 128 threads = 4 waves; each wave owns a 16x16 tile; 4 waves span
// the 64 output columns; one block handles 16 rows of x.
// A frag (16x4 f32): lanes 0-15 -> M=lane, v0/v1 = K,K+1; lanes 16-31 -> K+2,K+3
// B frag (4x16 f32): lanes 0-15 -> N=lane, v0/v1 = K,K+1; lanes 16-31 -> K+2,K+3
// D (16x16 f32): vgpr r -> M = r + 8*(lane>=16), N = lane%16
// ---------------------------------------------------------------------------
__global__ __launch_bounds__(128) void gemm1_wmma(const float* __restrict__ x,
                                                  const float* __restrict__ W1,
                                                  float* __restrict__ h,
                                                  int n) {
    const int wave   = threadIdx.x >> 5;
    const int lane   = threadIdx.x & 31;
    const int lr     = lane & 15;
    const int hi     = lane >> 4;                 // 0 or 1
    const int rowBase = blockIdx.x * 16;
    const int colBase = wave * 16;                // 4 waves -> 64 cols

    int arow = rowBase + lr; if (arow >= n) arow = n - 1;   // clamp: EXEC stays full for WMMA
    const int bcol = colBase + lr;
    const float* __restrict__ xrow = x + (long long)arow * D_IN;

    v8f c = {};
    for (int k = 0; k < D_IN; k += 4) {
        const int kk = k + 2 * hi;
        v2f a, b;
        a.x = xrow[kk];
        a.y = xrow[kk + 1];
        b.x = W1[(long long)kk       * D_HID + bcol];
        b.y = W1[(long long)(kk + 1) * D_HID + bcol];
        c = __builtin_amdgcn_wmma_f32_16x16x4_f32(false, a, false, b,
                                                  (short)0, c, false, false);
    }

    float* __restrict__ dcol = h + (long long)(rowBase + 8 * hi) * D_HID + colBase + lr;
    if (rowBase + 16 <= n) {                      // uniform fast path: full tile, no exec churn
        #pragma unroll
        for (int r = 0; r < 8; ++r)
            dcol[(long long)r * D_HID] = c[r];
    } else {                                      // ragged last tile (never taken for N%16==0)
        #pragma unroll
        for (int r = 0; r < 8; ++r) {
            const int row = rowBase + r + 8 * hi;
            if (row < n) h[(long long)row * D_HID + colBase + lr] = c[r];
        }
    }
}

// ---------------------------------------------------------------------------
// SpMM d=64: acc1[dst[e],:] += val[e] * h[src[e],:]
// wave-per-nnz; lane covers 2 dims via float2 (coalesced 256B row gather);
// unsafeAtomicAdd -> native global_atomic_add_f32 resolved in L2.
// ---------------------------------------------------------------------------
__global__ __launch_bounds__(256) void spmm64(const int* __restrict__ src,
                                              const int* __restrict__ dst,
                                              const float* __restrict__ val,
                                              const float* __restrict__ dense,
                                              float* __restrict__ acc,
                                              int nnz) {
    const int e = blockIdx.x * 8 + (threadIdx.x >> 5);
    if (e >= nnz) return;
    const int lane = threadIdx.x & 31;
    const int s = src[e];
    const int d = dst[e];
    const float v = val[e];
    const float2 m = ((const float2*)(dense + (long long)s * D_HID))[lane];
    float* out = acc + (long long)d * D_HID + 2 * lane;
    unsafeAtomicAdd(out,     v * m.x);
    unsafeAtomicAdd(out + 1, v * m.y);
}

// ---------------------------------------------------------------------------
// GEMM2: h2[N,32] = relu(acc1 + b1) @ W2[64,32]   (relu+bias fused into A load)
// block = 64 threads = 2 waves spanning the 32 output columns.
// ---------------------------------------------------------------------------
__global__ __launch_bounds__(64) void gemm2_wmma(const float* __restrict__ acc1,
                                                 const float* __restrict__ b1,
                                                 const float* __restrict__ W2,
                                                 float* __restrict__ h2,
                                                 int n) {
    const int wave   = threadIdx.x >> 5;
    const int lane   = threadIdx.x & 31;
    const int lr     = lane & 15;
    const int hi     = lane >> 4;
    const int rowBase = blockIdx.x * 16;
    const int colBase = wave * 16;                // 2 waves -> 32 cols

    int arow = rowBase + lr; if (arow >= n) arow = n - 1;
    const int bcol = colBase + lr;
    const float* __restrict__ zrow = acc1 + (long long)arow * D_HID;

    v8f c = {};
    for (int k = 0; k < D_HID; k += 4) {
        const int kk = k + 2 * hi;
        v2f a, b;
        float a0 = zrow[kk]     + b1[kk];
        float a1 = zrow[kk + 1] + b1[kk + 1];
        a.x = a0 > 0.0f ? a0 : 0.0f;              // fused ReLU
        a.y = a1 > 0.0f ? a1 : 0.0f;
        b.x = W2[(long long)kk       * D_EMB + bcol];
        b.y = W2[(long long)(kk + 1) * D_EMB + bcol];
        c = __builtin_amdgcn_wmma_f32_16x16x4_f32(false, a, false, b,
                                                  (short)0, c, false, false);
    }

    float* __restrict__ dcol = h2 + (long long)(rowBase + 8 * hi) * D_EMB + colBase + lr;
    if (rowBase + 16 <= n) {
        #pragma unroll
        for (int r = 0; r < 8; ++r)
            dcol[(long long)r * D_EMB] = c[r];
    } else {
        #pragma unroll
        for (int r = 0; r < 8; ++r) {
            const int row = rowBase + r + 8 * hi;
            if (row < n) h2[(long long)row * D_EMB + colBase + lr] = c[r];
        }
    }
}

// ---------------------------------------------------------------------------
// SpMM d=32: acc2[dst[e],:] += val[e] * h2[src[e],:] ; wave-per-nnz, lane-per-dim
// ---------------------------------------------------------------------------
__global__ __launch_bounds__(256) void spmm32(const int* __restrict__ src,
                                              const int* __restrict__ dst,
                                              const float* __restrict__ val,
                                              const float* __restrict__ dense,
                                              float* __restrict__ acc,
                                              int nnz) {
    const int e = blockIdx.x * 8 + (threadIdx.x >> 5);
    if (e >= nnz) return;
    const int lane = threadIdx.x & 31;
    const int s = src[e];
    const int d = dst[e];
    const float v = val[e];
    const float m = dense[(long long)s * D_EMB + lane];
    unsafeAtomicAdd(acc + (long long)d * D_EMB + lane, v * m);
}

// ---------------------------------------------------------------------------
// Decoder: scores[e] = dot(acc2[src]+b2, acc2[dst]+b2) ; wave-per-edge,
// lane-per-dim, wave32 xor-shuffle reduction, lane 0 stores.
// ---------------------------------------------------------------------------
__global__ __launch_bounds__(256) void decode(const int* __restrict__ esrc,
                                              const int* __restrict__ edst,
                                              const float* __restrict__ acc2,
                                              const float* __restrict__ b2,
                                              float* __restrict__ out,
                                              int nE) {
    const int e = blockIdx.x * 8 + (threadIdx.x >> 5);
    if (e >= nE) return;
    const int lane = threadIdx.x & 31;
    const int s = esrc[e];
    const int d = edst[e];
    const float bb = b2[lane];
    const float zs = acc2[(long long)s * D_EMB + lane] + bb;
    const float zd = acc2[(long long)d * D_EMB + lane] + bb;
    float p = zs * zd;
    #pragma unroll
    for (int off = 16; off > 0; off >>= 1)
        p += __shfl_xor(p, off, 32);
    if (lane == 0) out[e] = p;
}

// ---------------------------------------------------------------------------
extern "C" void kernel_launch(void* const* d_in, const int* in_sizes, int n_in,
                              void* d_out, int out_size, void* d_ws, size_t ws_size,
                              hipStream_t stream) {
    const float* x        = (const float*)d_in[0];
    const int*   adj_src  = (const int*)  d_in[1];
    const int*   adj_dst  = (const int*)  d_in[2];
    const float* adj_val  = (const float*)d_in[3];
    const int*   edge_idx = (const int*)  d_in[4];
    const float* W1       = (const float*)d_in[5];
    const float* b1       = (const float*)d_in[6];
    const float* W2       = (const float*)d_in[7];
    const float* b2       = (const float*)d_in[8];

    const int N   = in_sizes[0] / D_IN;      // 100000
    const int nnz = in_sizes[1];             // 3.2M
    const int nE  = in_sizes[4] / 2;         // 2M
    const int* esrc = edge_idx;
    const int* edst = edge_idx + nE;

    // workspace layout: h[N*64] | h2[N*32] | acc1[N*64] | acc2[N*32]
    float* h    = (float*)d_ws;
    float* h2   = h    + (long long)N * D_HID;
    float* acc1 = h2   + (long long)N * D_EMB;
    float* acc2 = acc1 + (long long)N * D_HID;

    // 1) zero acc1+acc2 (contiguous N*96 floats, float4-vectorized)
    {
        long long n4 = (long long)N * (D_HID + D_EMB) / 4;
        int blocks = (int)((n4 + 255) / 256);
        fill_zero4<<<blocks, 256, 0, stream>>>((float4*)acc1, n4);
    }
    // 2) h = x @ W1 (WMMA fp32)
    {
        int blocks = (N + 15) / 16;
        gemm1_wmma<<<blocks, 128, 0, stream>>>(x, W1, h, N);
    }
    // 3) acc1 += SpMM(A, h)
    {
        int blocks = (nnz + 7) / 8;
        spmm64<<<blocks, 256, 0, stream>>>(adj_src, adj_dst, adj_val, h, acc1, nnz);
    }
    // 4) h2 = relu(acc1 + b1) @ W2 (WMMA fp32, fused activation)
    {
        int blocks = (N + 15) / 16;
        gemm2_wmma<<<blocks, 64, 0, stream>>>(acc1, b1, W2, h2, N);
    }
    // 5) acc2 += SpMM(A, h2)
    {
        int blocks = (nnz + 7) / 8;
        spmm32<<<blocks, 256, 0, stream>>>(adj_src, adj_dst, adj_val, h2, acc2, nnz);
    }
    // 6) scores = rowwise dot over edges (fused + b2)
    {
        int blocks = (nE + 7) / 8;
        decode<<<blocks, 256, 0, stream>>>(esrc, edst, acc2, b2, (float*)d_out, nE);
    }
}